// EdgeConvNoCGN_82721070120986
// MI455X (gfx1250) — compile-verified
//
#include <hip/hip_runtime.h>
#include <math.h>

typedef float v2f __attribute__((ext_vector_type(2)));
typedef float v8f __attribute__((ext_vector_type(8)));

#define B_    4
#define CIN   64
#define COUT  64
#define NPTS  16384
#define KNN   16
#define NGRP  4
#define CPG   16          // channels per group
#define EPS_  1e-5f

// ------------------------------------------------------------------
// Zero the GroupNorm stats accumulators (2 floats per (b, group)).
// ------------------------------------------------------------------
__global__ void k_init(float* __restrict__ stats) {
    if (threadIdx.x < B_ * NGRP * 2) stats[threadIdx.x] = 0.0f;
}

// ------------------------------------------------------------------
// Fused dual GEMM with fp32 WMMA:
//   local = W1 @ feature,  edge = W2 @ feature   (per batch)
// Stored TRANSPOSED as [B, N, C] so that KNN gathers later read
// one contiguous 256B column per point.
// Each wave: one 16-wide N tile, all 4 M tiles, both weight mats.
// ------------------------------------------------------------------
__global__ __launch_bounds__(256) void k_gemm(const float* __restrict__ feat,
                                              const float* __restrict__ W1,
                                              const float* __restrict__ W2,
                                              float* __restrict__ local_t,
                                              float* __restrict__ edge_t) {
    __shared__ float sW[2 * COUT * CIN];   // 32 KB: W1 then W2, row-major [o][k]
    const int tid = threadIdx.x;
    for (int i = tid; i < COUT * CIN; i += 256) {
        sW[i]               = W1[i];
        sW[COUT * CIN + i]  = W2[i];
    }
    __syncthreads();

    const int wave = tid >> 5;
    const int lane = tid & 31;
    const int ln   = lane & 15;     // N (and M-row) index within tile
    const int hi   = lane >> 4;     // half-wave selector (K pairs / M+8)
    const int b    = blockIdx.x >> 7;                        // 128 blocks / batch
    const int tile = ((blockIdx.x & 127) << 3) + wave;       // n-tile in batch
    const int n0   = tile << 4;

    v8f acc[8];                     // [mat*4 + mtile]
#pragma unroll
    for (int i = 0; i < 8; ++i) acc[i] = (v8f){0, 0, 0, 0, 0, 0, 0, 0};

    const size_t fbase = (size_t)b * CIN * NPTS + (size_t)(n0 + ln);
    for (int kk = 0; kk < CIN; kk += 4) {
        const int k0 = kk + 2 * hi;            // this half-wave's K pair
        v2f bf;                                // B frag: 4x16 slice of feature
        bf.x = feat[fbase + (size_t)k0 * NPTS];
        bf.y = feat[fbase + (size_t)(k0 + 1) * NPTS];
#pragma unroll
        for (int mat = 0; mat < 2; ++mat) {
#pragma unroll
            for (int mt = 0; mt < 4; ++mt) {
                const float* w = &sW[mat * (COUT * CIN) + (mt * 16 + ln) * CIN + k0];
                v2f af;                        // A frag: 16x4 slice of W
                af.x = w[0];
                af.y = w[1];
                acc[mat * 4 + mt] = __builtin_amdgcn_wmma_f32_16x16x4_f32(
                    false, af, false, bf, (short)0, acc[mat * 4 + mt], false, false);
            }
        }
    }

    // D layout: VGPR r holds row M = mt*16 + r + 8*hi, col N = n0 + ln.
    // Store transposed [b][n][c]: rows r are consecutive channels -> float4 x2.
#pragma unroll
    for (int mat = 0; mat < 2; ++mat) {
        float* dst = mat ? edge_t : local_t;
#pragma unroll
        for (int mt = 0; mt < 4; ++mt) {
            v8f c = acc[mat * 4 + mt];
            size_t base = ((size_t)b * NPTS + (size_t)(n0 + ln)) * COUT
                          + (size_t)(mt * 16 + 8 * hi);
            *(float4*)(dst + base)     = make_float4(c[0], c[1], c[2], c[3]);
            *(float4*)(dst + base + 4) = make_float4(c[4], c[5], c[6], c[7]);
        }
    }
}

// ------------------------------------------------------------------
// Pass 1 over ef = gather(edge) - local: per-(b,group) sum & sumsq.
// Thread owns (n, channel-quad); 16 lanes with the same n read one
// full 256B column coalesced. LDS ds_add_f32 + global atomics.
// ------------------------------------------------------------------
__global__ __launch_bounds__(256) void k_stats(const int* __restrict__ knn,
                                               const float* __restrict__ local_t,
                                               const float* __restrict__ edge_t,
                                               float* __restrict__ stats) {
    __shared__ float sh[2 * NGRP];
    const int tid = threadIdx.x;
    if (tid < 2 * NGRP) sh[tid] = 0.0f;
    __syncthreads();

    const int b  = blockIdx.x >> 10;                 // 1024 blocks / batch
    const int t  = ((blockIdx.x & 1023) << 8) + tid; // 0 .. N*16-1
    const int c4 = t & 15;                           // channel quad 0..15
    const int n  = t >> 4;
    const int g  = c4 >> 2;                          // group 0..3

    const size_t colL = ((size_t)b * NPTS + n) * COUT + (size_t)c4 * 4;
    const float4 l4 = *(const float4*)(local_t + colL);
    const int* kp = knn + ((size_t)b * NPTS + n) * KNN;

    float s = 0.0f, ss = 0.0f;
#pragma unroll
    for (int j = 0; j < KNN; ++j) {
        const int idx = kp[j];
        const float4 e4 = *(const float4*)(edge_t + ((size_t)b * NPTS + idx) * COUT
                                           + (size_t)c4 * 4);
        const float d0 = e4.x - l4.x, d1 = e4.y - l4.y;
        const float d2 = e4.z - l4.z, d3 = e4.w - l4.w;
        s  += d0 + d1 + d2 + d3;
        ss += d0 * d0 + d1 * d1 + d2 * d2 + d3 * d3;
    }
    atomicAdd(&sh[g * 2 + 0], s);    // ds_add_f32
    atomicAdd(&sh[g * 2 + 1], ss);
    __syncthreads();
    if (tid < 2 * NGRP) atomicAdd(&stats[b * 2 * NGRP + tid], sh[tid]);
}

// ------------------------------------------------------------------
// Pass 2: normalize + affine + ReLU + mean over K, write [B,C,N].
// ------------------------------------------------------------------
__global__ __launch_bounds__(256) void k_out(const int* __restrict__ knn,
                                             const float* __restrict__ local_t,
                                             const float* __restrict__ edge_t,
                                             const float* __restrict__ stats,
                                             const float* __restrict__ gw,
                                             const float* __restrict__ gb,
                                             float* __restrict__ out) {
    const int tid = threadIdx.x;
    const int b  = blockIdx.x >> 10;
    const int t  = ((blockIdx.x & 1023) << 8) + tid;
    const int c4 = t & 15;
    const int n  = t >> 4;
    const int g  = c4 >> 2;

    const float inv_cnt = 1.0f / (float)((size_t)CPG * NPTS * KNN);
    const float mean = stats[b * 2 * NGRP + g * 2 + 0] * inv_cnt;
    const float msq  = stats[b * 2 * NGRP + g * 2 + 1] * inv_cnt;
    const float inv  = rsqrtf(msq - mean * mean + EPS_);

    const float4 gamma = ((const float4*)gw)[c4];
    const float4 beta  = ((const float4*)gb)[c4];
    const float4 l4 = *(const float4*)(local_t + ((size_t)b * NPTS + n) * COUT
                                       + (size_t)c4 * 4);
    const int* kp = knn + ((size_t)b * NPTS + n) * KNN;

    float a0 = 0.0f, a1 = 0.0f, a2 = 0.0f, a3 = 0.0f;
#pragma unroll
    for (int j = 0; j < KNN; ++j) {
        const int idx = kp[j];
        const float4 e4 = *(const float4*)(edge_t + ((size_t)b * NPTS + idx) * COUT
                                           + (size_t)c4 * 4);
        float v0 = ((e4.x - l4.x) - mean) * inv * gamma.x + beta.x;
        float v1 = ((e4.y - l4.y) - mean) * inv * gamma.y + beta.y;
        float v2 = ((e4.z - l4.z) - mean) * inv * gamma.z + beta.z;
        float v3 = ((e4.w - l4.w) - mean) * inv * gamma.w + beta.w;
        a0 += fmaxf(v0, 0.0f);
        a1 += fmaxf(v1, 0.0f);
        a2 += fmaxf(v2, 0.0f);
        a3 += fmaxf(v3, 0.0f);
    }
    const float r = 1.0f / (float)KNN;
    const size_t ob = ((size_t)b * COUT + (size_t)c4 * 4) * NPTS + (size_t)n;
    out[ob]                     = a0 * r;
    out[ob + (size_t)NPTS]      = a1 * r;
    out[ob + 2 * (size_t)NPTS]  = a2 * r;
    out[ob + 3 * (size_t)NPTS]  = a3 * r;
}

// ------------------------------------------------------------------
extern "C" void kernel_launch(void* const* d_in, const int* in_sizes, int n_in,
                              void* d_out, int out_size, void* d_ws, size_t ws_size,
                              hipStream_t stream) {
    (void)in_sizes; (void)n_in; (void)out_size; (void)ws_size;

    const float* feat = (const float*)d_in[0];   // [B, CIN, N]
    const int*   knn  = (const int*)d_in[1];     // [B, N, K]
    const float* W1   = (const float*)d_in[2];   // [COUT, CIN]
    const float* W2   = (const float*)d_in[3];   // [COUT, CIN]
    const float* gw   = (const float*)d_in[4];   // [COUT]
    const float* gb   = (const float*)d_in[5];   // [COUT]
    float*       out  = (float*)d_out;           // [B, COUT, N]

    float* local_t = (float*)d_ws;                              // [B, N, COUT] 16.8 MB
    float* edge_t  = local_t + (size_t)B_ * NPTS * COUT;        // [B, N, COUT] 16.8 MB
    float* stats   = edge_t  + (size_t)B_ * NPTS * COUT;        // [B, NGRP, 2]

    k_init <<<1, 64, 0, stream>>>(stats);
    k_gemm <<<B_ * (NPTS / 16) / 8, 256, 0, stream>>>(feat, W1, W2, local_t, edge_t);
    k_stats<<<B_ * (NPTS * KNN) / 256, 256, 0, stream>>>(knn, local_t, edge_t, stats);
    k_out  <<<B_ * (NPTS * KNN) / 256, 256, 0, stream>>>(knn, local_t, edge_t, stats,
                                                         gw, gb, out);
}